// CONVMGEmbedder_21062519620310
// MI455X (gfx1250) — compile-verified
//
#include <hip/hip_runtime.h>

#define N_NODES 100000
#define N_EDGES 1600000
#define C_DIM   128
#define B_GR    16
#define L_LAYERS 3
#define EPS_F   1e-5f
#define SLOPE_F 0.2f

typedef float v2f __attribute__((ext_vector_type(2)));
typedef float v8f __attribute__((ext_vector_type(8)));
typedef unsigned int u32x4 __attribute__((ext_vector_type(4)));
typedef int i32x4 __attribute__((ext_vector_type(4)));
typedef int i32x8 __attribute__((ext_vector_type(8)));

#if __has_builtin(__builtin_amdgcn_tensor_load_to_lds)
#define HAVE_TDM 1
#else
#define HAVE_TDM 0
#endif

__device__ __forceinline__ float leaky(float x) { return x >= 0.f ? x : SLOPE_F * x; }

// ---------------- utility: zero fill (graph-capture safe, no memset) ----------
__global__ void k_zero(float* __restrict__ p, long n) {
    long i = (long)blockIdx.x * blockDim.x + threadIdx.x;
    long stride = (long)gridDim.x * blockDim.x;
    for (; i < n; i += stride) p[i] = 0.f;
}

// ---------------- degrees ----------------------------------------------------
__global__ void k_degrees(const int* __restrict__ src, const int* __restrict__ dst,
                          float* __restrict__ degO, float* __restrict__ degI) {
    int i = blockIdx.x * blockDim.x + threadIdx.x;
    int stride = gridDim.x * blockDim.x;
    for (; i < N_EDGES; i += stride) {
        atomicAdd(&degO[src[i]], 1.f);
        atomicAdd(&degI[dst[i]], 1.f);
    }
}

__global__ void k_gcnt(const int* __restrict__ gid, float* __restrict__ gcnt) {
    int i = blockIdx.x * blockDim.x + threadIdx.x;
    if (i < N_NODES) atomicAdd(&gcnt[gid[i]], 1.f);
}

__global__ void k_invsqrt(float* __restrict__ degO, float* __restrict__ degI) {
    int i = blockIdx.x * blockDim.x + threadIdx.x;
    if (i < N_NODES) {
        float a = degO[i]; degO[i] = rsqrtf(a < 1.f ? 1.f : a);
        float b = degI[i]; degI[i] = rsqrtf(b < 1.f ? 1.f : b);
    }
}

// ---------------- edge scatter: agg[dst] += h[src] * invO[src] ---------------
// One wave32 per edge; each lane moves a float4 (32 lanes * 4 = 128 channels).
// Gathers and atomics stay L2-resident (51 MB table << 192 MB L2).
__global__ __launch_bounds__(256) void k_scatter(const float* __restrict__ h,
                                                 const float* __restrict__ invO,
                                                 const int* __restrict__ src,
                                                 const int* __restrict__ dst,
                                                 float* __restrict__ agg) {
    int wid  = threadIdx.x >> 5;
    int lane = threadIdx.x & 31;
    int e = blockIdx.x * 8 + wid;
    if (e >= N_EDGES) return;
    int s = src[e], d = dst[e];
    float sc = invO[s];
    float4 v = ((const float4*)(h + (size_t)s * C_DIM))[lane];
    v.x *= sc; v.y *= sc; v.z *= sc; v.w *= sc;
    float* ap = agg + (size_t)d * C_DIM + lane * 4;
    atomicAdd(ap + 0, v.x);
    atomicAdd(ap + 1, v.y);
    atomicAdd(ap + 2, v.z);
    atomicAdd(ap + 3, v.w);
}

// ---------------- GEMM: out = (agg * invI) @ W, f32 WMMA 16x16x4 -------------
// Block = 128 threads = 4 waves; each wave owns 16 rows x 128 cols of output.
// A tile (64x128 f32) DMA'd into LDS by the Tensor Data Mover with +4-DWORD
// row padding (stride 132 -> conflict-free ds_load_b64 fragment reads).
#define TILE_ROWS  64
#define LDS_STRIDE 132

#if HAVE_TDM
// D# descriptor for a 2D tile: tile_dim = 128 x 64 f32, row stride 128,
// pad 4 DWORDs every 128 DWORDs, OOB rows (>= rows_avail) read as zero.
__device__ __forceinline__ void tdm_load_tile(const float* gptr, unsigned lds_byte_off,
                                              unsigned rows_avail) {
    unsigned long long ga = (unsigned long long)(uintptr_t)gptr;
    u32x4 g0;
    g0.x = 1u;                                        // count=1 (valid), user mode
    g0.y = lds_byte_off;                              // lds_addr
    g0.z = (unsigned)(ga & 0xFFFFFFFFu);              // global_addr[31:0]
    g0.w = (unsigned)((ga >> 32) & 0x01FFFFFFu)       // global_addr[56:32]
           | (2u << 30);                              // type = 2 ("image")
    i32x8 g1;
    g1[0] = (int)((2u << 16)      // data_size = 4 bytes
                | (1u << 20)      // pad_enable
                | (6u << 22)      // pad_interval: 128 DWORDs
                | (3u << 25));    // pad_amount: 4 DWORDs
    g1[1] = (int)(128u << 16);                        // tensor_dim0 = 128 (lo16)
    g1[2] = (int)((rows_avail & 0xFFFFu) << 16);      // dim0 hi=0 | tensor_dim1 lo16
    g1[3] = (int)((rows_avail >> 16) | (128u << 16)); // tensor_dim1 hi | tile_dim0=128
    g1[4] = (int)TILE_ROWS;                           // tile_dim1 = 64, tile_dim2 = 0
    g1[5] = (int)C_DIM;                               // tensor_dim0_stride = 128 (lo32)
    g1[6] = 0;
    g1[7] = 0;
    i32x4 z4 = {0, 0, 0, 0};
    i32x8 z8 = {0, 0, 0, 0, 0, 0, 0, 0};
    __builtin_amdgcn_tensor_load_to_lds(g0, g1, z4, z4, z8, 0);
}
#endif

__global__ __launch_bounds__(128) void k_gemm(const float* __restrict__ agg,
                                              const float* __restrict__ invI,
                                              const float* __restrict__ Wl,
                                              float* __restrict__ out) {
    __shared__ float As[TILE_ROWS * LDS_STRIDE];
    int tid = threadIdx.x;
    int r0 = blockIdx.x * TILE_ROWS;

#if HAVE_TDM
    if (tid < 32) {  // one wave issues the tensor DMA for the whole block
        tdm_load_tile(agg + (size_t)r0 * C_DIM,
                      (unsigned)(uintptr_t)&As[0],
                      (unsigned)(N_NODES - r0));
        __builtin_amdgcn_s_wait_tensorcnt(0);
    }
    __syncthreads();
#else
    for (int idx = tid; idx < TILE_ROWS * C_DIM; idx += 128) {
        int r = idx >> 7, c = idx & 127;
        int row = r0 + r;
        As[r * LDS_STRIDE + c] = (row < N_NODES) ? agg[(size_t)row * C_DIM + c] : 0.f;
    }
    __syncthreads();
#endif

    int wid = tid >> 5, lane = tid & 31;
    int lm = lane & 15, lh = lane >> 4;
    int m0 = wid * 16;

    v8f acc[8];
    v8f zero = {};
    #pragma unroll
    for (int nt = 0; nt < 8; nt++) acc[nt] = zero;

    const int arow = m0 + lm;                 // A-matrix row held by this lane
    int grow = r0 + arow;
    float si = invI[grow < N_NODES ? grow : (N_NODES - 1)];  // D_in^-1/2 row scale

    for (int kk = 0; kk < 32; kk++) {
        // A 16x4 f32 fragment: v0=K0,v1=K1 (lanes 0-15), K2,K3 (lanes 16-31)
        int k0 = kk * 4 + lh * 2;
        v2f a = *(const v2f*)&As[arow * LDS_STRIDE + k0];
        a.x *= si; a.y *= si;
        #pragma unroll
        for (int nt = 0; nt < 8; nt++) {
            int col = nt * 16 + lm;
            v2f b;  // B 4x16: VGPR v holds row K=v (lanes 0-15) / K=v+2 (16-31)
            b.x = Wl[(size_t)(k0 + 0) * C_DIM + col];
            b.y = Wl[(size_t)(k0 + 1) * C_DIM + col];
            acc[nt] = __builtin_amdgcn_wmma_f32_16x16x4_f32(
                false, a, false, b, (short)0, acc[nt], false, false);
        }
    }

    // D layout: VGPR r -> M=r (lanes 0-15) / M=r+8 (lanes 16-31)
    if (r0 + TILE_ROWS <= N_NODES) {   // uniform branch: full tile, no guards
        #pragma unroll
        for (int nt = 0; nt < 8; nt++) {
            int col = nt * 16 + lm;
            #pragma unroll
            for (int r = 0; r < 8; r++) {
                int row = r0 + m0 + r + 8 * lh;
                out[(size_t)row * C_DIM + col] = acc[nt][r];
            }
        }
    } else {
        #pragma unroll
        for (int nt = 0; nt < 8; nt++) {
            int col = nt * 16 + lm;
            #pragma unroll
            for (int r = 0; r < 8; r++) {
                int row = r0 + m0 + r + 8 * lh;
                if (row < N_NODES) out[(size_t)row * C_DIM + col] = acc[nt][r];
            }
        }
    }
}

// ---------------- graph segment sums via hierarchical LDS atomics ------------
__global__ __launch_bounds__(256) void k_stats(const float* __restrict__ h,
                                               const int* __restrict__ gid,
                                               float* __restrict__ gs,
                                               float* __restrict__ gss) {
    __shared__ float s_sum[B_GR * C_DIM];
    __shared__ float s_sq[B_GR * C_DIM];
    int tid = threadIdx.x;
    for (int i = tid; i < B_GR * C_DIM; i += 256) { s_sum[i] = 0.f; s_sq[i] = 0.f; }
    __syncthreads();

    int c = tid & 127, rg = tid >> 7;
    int r0 = blockIdx.x * 256;
    int rend = r0 + 256; if (rend > N_NODES) rend = N_NODES;
    for (int r = r0 + rg; r < rend; r += 2) {
        float x = h[(size_t)r * C_DIM + c];
        int g = gid[r];
        atomicAdd(&s_sum[g * C_DIM + c], x);
        atomicAdd(&s_sq[g * C_DIM + c], x * x);
    }
    __syncthreads();
    for (int i = tid; i < B_GR * C_DIM; i += 256) {
        float a = s_sum[i], b = s_sq[i];
        if (a != 0.f) atomicAdd(&gs[i], a);
        if (b != 0.f) atomicAdd(&gss[i], b);
    }
}

// ---------------- per-layer stat finalization --------------------------------
__global__ __launch_bounds__(256) void k_finalize(const float* __restrict__ gs,
                                                  const float* __restrict__ gss,
                                                  const float* __restrict__ gcnt,
                                                  const float* __restrict__ lam_l,
                                                  float* __restrict__ gm, float* __restrict__ grs,
                                                  float* __restrict__ bm, float* __restrict__ brs,
                                                  float* __restrict__ lamw) {
    int tid = threadIdx.x;
    for (int i = tid; i < B_GR * C_DIM; i += 256) {
        int g = i >> 7;
        float cnt = gcnt[g]; cnt = cnt < 1.f ? 1.f : cnt;
        float mu = gs[i] / cnt;
        float var = gss[i] / cnt - mu * mu;
        gm[i] = mu; grs[i] = rsqrtf(var + EPS_F);
    }
    if (tid < C_DIM) {
        float s = 0.f, q = 0.f;
        for (int g = 0; g < B_GR; g++) { s += gs[g * C_DIM + tid]; q += gss[g * C_DIM + tid]; }
        float mu = s / (float)N_NODES;
        float var = q / (float)N_NODES - mu * mu;
        bm[tid] = mu; brs[tid] = rsqrtf(var + EPS_F);
    }
    if (tid == 0) {
        float l0 = lam_l[0], l1 = lam_l[1], l2 = lam_l[2];
        float m = fmaxf(l0, fmaxf(l1, l2));
        float e0 = __expf(l0 - m), e1 = __expf(l1 - m), e2 = __expf(l2 - m);
        float inv = 1.f / (e0 + e1 + e2);
        lamw[0] = e0 * inv; lamw[1] = e1 * inv; lamw[2] = e2 * inv;
    }
}

// ---------------- UnitedNorm + LeakyReLU + (optional) readout accum ----------
// One wave per node; node mean/var via wave32 shuffle reduction.
__global__ __launch_bounds__(256) void k_norm(float* __restrict__ h,
                                              const int* __restrict__ gid,
                                              const float* __restrict__ bm, const float* __restrict__ brs,
                                              const float* __restrict__ gm, const float* __restrict__ grs,
                                              const float* __restrict__ gamma_l,
                                              const float* __restrict__ beta_l,
                                              const float* __restrict__ lamw,
                                              float* __restrict__ emb, int do_emb) {
    int wid = threadIdx.x >> 5, lane = threadIdx.x & 31;
    int node = blockIdx.x * 8 + wid;
    if (node >= N_NODES) return;
    float w0 = lamw[0], w1 = lamw[1], w2 = lamw[2];
    int g = gid[node];
    float4 x = *(const float4*)(h + (size_t)node * C_DIM + lane * 4);
    float s = x.x + x.y + x.z + x.w;
    float q = x.x * x.x + x.y * x.y + x.z * x.z + x.w * x.w;
    for (int off = 16; off; off >>= 1) {
        s += __shfl_xor(s, off, 32);
        q += __shfl_xor(q, off, 32);
    }
    const float invC = 1.f / (float)C_DIM;
    float nm = s * invC;
    float nv = q * invC - nm * nm;
    float nrs = rsqrtf(nv + EPS_F);

    float xv[4] = {x.x, x.y, x.z, x.w};
    float ov[4];
    int c0 = lane * 4;
    #pragma unroll
    for (int j = 0; j < 4; j++) {
        int c = c0 + j;
        float xx = xv[j];
        float bn = (xx - bm[c]) * brs[c];
        float gn = (xx - gm[g * C_DIM + c]) * grs[g * C_DIM + c];
        float nn = (xx - nm) * nrs;
        float y = gamma_l[c] * (w0 * bn + w1 * gn + w2 * nn) + beta_l[c];
        y = leaky(y);
        ov[j] = y;
        if (do_emb) atomicAdd(&emb[g * C_DIM + c], y);
    }
    float4 o = {ov[0], ov[1], ov[2], ov[3]};
    *(float4*)(h + (size_t)node * C_DIM + lane * 4) = o;
}

// ---------------- final readout ----------------------------------------------
__global__ void k_readout(const float* __restrict__ emb, const float* __restrict__ gcnt,
                          float* __restrict__ out) {
    int i = blockIdx.x * blockDim.x + threadIdx.x;
    if (i < B_GR * C_DIM) {
        int g = i >> 7;
        float cnt = gcnt[g]; cnt = cnt < 1.f ? 1.f : cnt;
        out[i] = leaky(emb[i] / cnt);
    }
}

extern "C" void kernel_launch(void* const* d_in, const int* in_sizes, int n_in,
                              void* d_out, int out_size, void* d_ws, size_t ws_size,
                              hipStream_t stream) {
    (void)in_sizes; (void)n_in; (void)out_size; (void)ws_size;
    const float* node_feats = (const float*)d_in[0];
    const float* Wmat       = (const float*)d_in[1];
    const float* gamma      = (const float*)d_in[2];
    const float* beta       = (const float*)d_in[3];
    const float* lambdas    = (const float*)d_in[4];
    const int*   src        = (const int*)d_in[5];
    const int*   dst        = (const int*)d_in[6];
    const int*   gid        = (const int*)d_in[7];
    float* out = (float*)d_out;

    float* ws = (float*)d_ws;
    const size_t NC = (size_t)N_NODES * C_DIM;
    float* agg  = ws;
    float* hbuf = ws + NC;
    float* invO = ws + 2 * NC;
    float* invI = invO + N_NODES;
    float* gcnt = invI + N_NODES;
    float* gs   = gcnt + B_GR;
    float* gss  = gs  + B_GR * C_DIM;
    float* gm   = gss + B_GR * C_DIM;
    float* grs  = gm  + B_GR * C_DIM;
    float* bm   = grs + B_GR * C_DIM;
    float* brs  = bm  + C_DIM;
    float* emb  = brs + C_DIM;
    float* lamw = emb + B_GR * C_DIM;
    const long small_n = (long)(4 * B_GR * C_DIM + 2 * C_DIM + B_GR * C_DIM + 4); // gs..lamw

    // degrees + graph counts (recomputed every call; deterministic)
    k_zero<<<512, 256, 0, stream>>>(invO, (long)(2 * N_NODES + B_GR));
    k_degrees<<<1024, 256, 0, stream>>>(src, dst, invO, invI);
    k_gcnt<<<(N_NODES + 255) / 256, 256, 0, stream>>>(gid, gcnt);
    k_invsqrt<<<(N_NODES + 255) / 256, 256, 0, stream>>>(invO, invI);

    const float* hcur = node_feats;
    for (int l = 0; l < L_LAYERS; l++) {
        k_zero<<<2048, 256, 0, stream>>>(agg, (long)NC);
        k_zero<<<16, 256, 0, stream>>>(gs, small_n);
        k_scatter<<<(N_EDGES + 7) / 8, 256, 0, stream>>>(hcur, invO, src, dst, agg);
        k_gemm<<<(N_NODES + TILE_ROWS - 1) / TILE_ROWS, 128, 0, stream>>>(
            agg, invI, Wmat + (size_t)l * C_DIM * C_DIM, hbuf);
        k_stats<<<(N_NODES + 255) / 256, 256, 0, stream>>>(hbuf, gid, gs, gss);
        k_finalize<<<1, 256, 0, stream>>>(gs, gss, gcnt, lambdas + l * 3,
                                          gm, grs, bm, brs, lamw);
        k_norm<<<(N_NODES + 7) / 8, 256, 0, stream>>>(
            hbuf, gid, bm, brs, gm, grs, gamma + l * C_DIM, beta + l * C_DIM,
            lamw, emb, (l == L_LAYERS - 1) ? 1 : 0);
        hcur = hbuf;
    }
    k_readout<<<(B_GR * C_DIM + 255) / 256, 256, 0, stream>>>(emb, gcnt, out);
}